// QuantumLayer_36575941493557
// MI455X (gfx1250) — compile-verified
//
#include <hip/hip_runtime.h>

// Quantum layer: 4 qubits, 2 layers, batch B. out[b,i] = <Z_i>.
// amp[b,:] = U * v(x_b) with U = fixed 16x16 complex unitary built per-block
// from weights; the batched GEMM runs as two chained v_wmma_f32_16x16x4_f32
// pipelines (Re/Im), K=16 via 4 slices. HBM-bound: ~25 MB -> ~1.1 us.
// Grid-stride (4 tiles/wave) amortizes the per-block U-builder 4x.

typedef __attribute__((ext_vector_type(2))) float v2f;
typedef __attribute__((ext_vector_type(8))) float v8f;

#define WAVES 8
#define TPB   256
#define TPW   4      // tiles per wave (grid-stride)

__global__ __launch_bounds__(TPB) void qlayer_kernel(const float* __restrict__ x,
                                                     const float* __restrict__ w,
                                                     float* __restrict__ out,
                                                     int B)
{
    // sW[0]=Re W, sW[1]=Im W, where W[k][n] = U[n][k] (GEMM B matrix, k*16+n)
    __shared__ float sW[2][256];
    __shared__ float sP[WAVES][256];   // per-wave probs staging: [sample][ampIndex]

    const int tid = threadIdx.x;

    // ---------------- build layer unitary U (16 threads, one column each) ---------
    if (tid < 16) {
        const int j = tid;
        float sr[16], si[16];
#pragma unroll
        for (int n = 0; n < 16; ++n) { sr[n] = (n == j) ? 1.0f : 0.0f; si[n] = 0.0f; }
#pragma unroll
        for (int l = 0; l < 2; ++l) {
#pragma unroll
            for (int q = 0; q < 4; ++q) {
                const float th0 = w[(l*4 + q)*2 + 0];
                const float th1 = w[(l*4 + q)*2 + 1];
                const int  mask = 1 << (3 - q);            // wire 0 = MSB
                // RZ(th0): diag(e^{-i t/2}, e^{+i t/2})
                float cz, sz; __sincosf(0.5f*th0, &sz, &cz);
#pragma unroll
                for (int n = 0; n < 16; ++n) {
                    const float wi = (n & mask) ? sz : -sz;
                    const float nr = sr[n]*cz - si[n]*wi;
                    const float ni = sr[n]*wi + si[n]*cz;
                    sr[n] = nr; si[n] = ni;
                }
                // RY(th1): [[c,-s],[s,c]]
                float cy, sy; __sincosf(0.5f*th1, &sy, &cy);
#pragma unroll
                for (int n = 0; n < 16; ++n) {
                    if (!(n & mask)) {
                        const int n1 = n | mask;
                        const float a0r = sr[n],  a0i = si[n];
                        const float a1r = sr[n1], a1i = si[n1];
                        sr[n]  = cy*a0r - sy*a1r;  si[n]  = cy*a0i - sy*a1i;
                        sr[n1] = sy*a0r + cy*a1r;  si[n1] = sy*a0i + cy*a1i;
                    }
                }
            }
            // CNOT ring (0,1),(1,2),(2,3),(3,0): for control=1 swap target pair
#pragma unroll
            for (int e = 0; e < 4; ++e) {
                const int cm = 1 << (3 - e);
                const int tm = 1 << (3 - ((e + 1) & 3));
#pragma unroll
                for (int n = 0; n < 16; ++n) {
                    if ((n & cm) && !(n & tm)) {
                        const int n1 = n | tm;
                        float t;
                        t = sr[n]; sr[n] = sr[n1]; sr[n1] = t;
                        t = si[n]; si[n] = si[n1]; si[n1] = t;
                    }
                }
            }
        }
        // column j of U -> W rows: W[j][n] = U[n][j]
#pragma unroll
        for (int n = 0; n < 16; ++n) { sW[0][j*16 + n] = sr[n]; sW[1][j*16 + n] = si[n]; }
    }
    __syncthreads();

    const int wave = tid >> 5;
    const int lane = tid & 31;
    const int m    = lane & 15;      // A row index / D column (amp) index
    const int h    = lane >> 4;      // half-wave

    // B-matrix regs (loaded once per wave):
    // f32 B 4x16 layout -> VGPR j, lane(n+16h) holds B[k = j+2h, n]
    v2f br[4], bi[4];
#pragma unroll
    for (int t = 0; t < 4; ++t) {
        const int k0 = 4*t + 2*h;
        br[t].x = sW[0][(k0+0)*16 + m]; br[t].y = sW[0][(k0+1)*16 + m];
        bi[t].x = sW[1][(k0+0)*16 + m]; bi[t].y = sW[1][(k0+1)*16 + m];
    }

    const int ntiles = (B + 15) >> 4;
    const int stride = gridDim.x * WAVES;
    float* pl = &sP[wave][0];

    for (int tile = blockIdx.x * WAVES + wave; tile < ntiles; tile += stride) {
        // ------------ encoded product state: only the 8 elems this lane feeds -----
        const float4 xv = ((const float4*)x)[tile*16 + m];
        float c0, s0, c1, s1, c2, s2, c3, s3;
        __sincosf(0.5f*xv.x, &s0, &c0);
        __sincosf(0.5f*xv.y, &s1, &c1);
        __sincosf(0.5f*xv.z, &s2, &c2);
        __sincosf(0.5f*xv.w, &s3, &c3);
        // lane needs v[k] with k = 4t + 2h + j:  v = p01[t] * t2[h] * t3[j]
        const float p01_0 = c0*c1, p01_1 = c0*s1, p01_2 = s0*c1, p01_3 = s0*s1;
        const float t2h = h ? s2 : c2;                 // single cndmask
        const float qc = t2h * c3, qs = t2h * s3;
        v2f aT[4];
        aT[0].x = p01_0*qc; aT[0].y = p01_0*qs;
        aT[1].x = p01_1*qc; aT[1].y = p01_1*qs;
        aT[2].x = p01_2*qc; aT[2].y = p01_2*qs;
        aT[3].x = p01_3*qc; aT[3].y = p01_3*qs;

        // ------------ amp = V * W via chained f32 WMMA (K=16 = 4 slices) ----------
        v8f cr = {}; v8f ci = {};
#pragma unroll
        for (int t = 0; t < 4; ++t) {
            cr = __builtin_amdgcn_wmma_f32_16x16x4_f32(false, aT[t], false, br[t], (short)0, cr, false, false);
            ci = __builtin_amdgcn_wmma_f32_16x16x4_f32(false, aT[t], false, bi[t], (short)0, ci, false, false);
        }

        // ------------ probs -> LDS [sample][amp] ----------------------------------
        // D layout: lane(n=m, h), VGPR r holds D[sample = r+8h, amp = m]
#pragma unroll
        for (int r = 0; r < 8; ++r) {
            const float p = cr[r]*cr[r] + ci[r]*ci[r];
            pl[(r + 8*h)*16 + m] = p;
        }
        asm volatile("s_wait_dscnt 0" ::: "memory");

        // ------------ Z expectations: Walsh-style signed reduction ----------------
        if (lane < 16) {
            const float4* pr = (const float4*)&pl[lane*16];     // 16 contiguous probs
            const float4 q0 = pr[0], q1 = pr[1], q2 = pr[2], q3 = pr[3];
            const float e0 = q0.x+q0.y, d0 = q0.x-q0.y;
            const float e1 = q0.z+q0.w, d1 = q0.z-q0.w;
            const float e2 = q1.x+q1.y, d2 = q1.x-q1.y;
            const float e3 = q1.z+q1.w, d3 = q1.z-q1.w;
            const float e4 = q2.x+q2.y, d4 = q2.x-q2.y;
            const float e5 = q2.z+q2.w, d5 = q2.z-q2.w;
            const float e6 = q3.x+q3.y, d6 = q3.x-q3.y;
            const float e7 = q3.z+q3.w, d7 = q3.z-q3.w;
            const float o3 = ((d0+d1)+(d2+d3)) + ((d4+d5)+(d6+d7));   // sign by bit0
            const float ee0 = e0+e1, ed0 = e0-e1;
            const float ee1 = e2+e3, ed1 = e2-e3;
            const float ee2 = e4+e5, ed2 = e4-e5;
            const float ee3 = e6+e7, ed3 = e6-e7;
            const float o2 = (ed0+ed1) + (ed2+ed3);                   // sign by bit1
            const float o1 = (ee0-ee1) + (ee2-ee3);                   // sign by bit2
            const float o0 = (ee0+ee1) - (ee2+ee3);                   // sign by bit3 (wire 0 = MSB)
            ((float4*)out)[tile*16 + lane] = make_float4(o0, o1, o2, o3);
        }
        // WAR on pl across iterations is safe: same-wave LDS ops are in-order.
    }
}

extern "C" void kernel_launch(void* const* d_in, const int* in_sizes, int n_in,
                              void* d_out, int out_size, void* d_ws, size_t ws_size,
                              hipStream_t stream) {
    const float* x = (const float*)d_in[0];   // [B,4]
    const float* w = (const float*)d_in[1];   // [2,4,2]
    float* out = (float*)d_out;               // [B,4]
    const int B = in_sizes[0] / 4;
    const int tiles = (B + 15) / 16;
    const int blocks = (tiles + WAVES*TPW - 1) / (WAVES*TPW);
    qlayer_kernel<<<blocks, TPB, 0, stream>>>(x, w, out, B);
}